// LightGCN_55430847922200
// MI455X (gfx1250) — compile-verified
//
#include <hip/hip_runtime.h>

#define TILE 1024            // edges per staging tile (NNZ=9.6M is a multiple)
#define WAVE_EDGES 128       // TILE / 8 waves
#define SPMM_BLOCK 256       // 8 wave32s
#define SPMM_GRID 1250
#define EW_BLOCK 256
#define EW_GRID 4096

// ---- CDNA5 async global->LDS copy (16B per lane), non-temporal so the edge
// ---- stream does not evict the L2-resident embedding tables.
__device__ __forceinline__ void async_copy16_nt(const void* g, void* l) {
    unsigned lds = (unsigned)(unsigned long long)l;   // low 32b of LDS flat addr == LDS offset
    unsigned long long ga = (unsigned long long)g;
    asm volatile("global_load_async_to_lds_b128 %0, %1, off th:TH_LOAD_NT"
                 :: "v"(lds), "v"(ga)
                 : "memory");
}
__device__ __forceinline__ void wait_async_le3() {
    asm volatile("s_wait_asynccnt 3" ::: "memory");
}
__device__ __forceinline__ void wait_async_le0() {
    asm volatile("s_wait_asynccnt 0" ::: "memory");
}
__device__ __forceinline__ void wait_ds0() {
    asm volatile("s_wait_dscnt 0" ::: "memory");
}

// ---------------- SpMM: out[r,:] += v * ego[c,:]  (COO, d=64) ----------------
// Each wave owns a private 128-edge slice of each LDS buffer: it stages its own
// slice (1 async b128 per array, 512B/wave) and processes only that slice, so
// no block barriers are needed -> 8 independent per-wave pipelines per block.
__global__ void __launch_bounds__(SPMM_BLOCK)
lg_spmm(const int* __restrict__ rows, const int* __restrict__ cols,
        const float* __restrict__ vals, const float* __restrict__ ego,
        float* __restrict__ outv, int nnz) {
    __shared__ __align__(16) int   sRows[2][TILE];
    __shared__ __align__(16) int   sCols[2][TILE];
    __shared__ __align__(16) float sVals[2][TILE];

    const int tid  = threadIdx.x;
    const int lane = tid & 31;
    const int wid  = tid >> 5;            // 0..7
    const int ntiles = nnz / TILE;
    const int sbase = (wid << 7) + (lane << 2);   // this lane's 4-int slot

    auto stage = [&](int tileIdx, int b) {
        const int off = tileIdx * TILE + sbase;
        async_copy16_nt(rows + off, &sRows[b][sbase]);
        async_copy16_nt(cols + off, &sCols[b][sbase]);
        async_copy16_nt(vals + off, &sVals[b][sbase]);
    };

    int t = (int)blockIdx.x;
    if (t < ntiles) {
        int buf = 0;
        stage(t, 0);                                      // prologue
        for (; t < ntiles; t += (int)gridDim.x) {
            const int tn = t + (int)gridDim.x;
            const bool more = (tn < ntiles);
            if (more) {                                   // overlap next tile's DMA
                wait_ds0();                               // prior reads of buf^1 fully retired
                stage(tn, buf ^ 1);
                wait_async_le3();                         // oldest 3 (current tile) landed
            } else {
                wait_async_le0();                         // drain
            }

            // one wave per edge: 32 lanes x float2 cover d=64.
            // Edge params are wave-uniform -> readfirstlane => SALU addressing,
            // saddr-form gather loads and atomics.
            const int ebase = wid << 7;
            #pragma unroll 4
            for (int e = 0; e < WAVE_EDGES; ++e) {
                const int r = __builtin_amdgcn_readfirstlane(sRows[buf][ebase + e]);
                const int c = __builtin_amdgcn_readfirstlane(sCols[buf][ebase + e]);
                const float v = __int_as_float(
                    __builtin_amdgcn_readfirstlane(__float_as_int(sVals[buf][ebase + e])));
                const float* __restrict__ srcRow = ego + ((size_t)(unsigned)c << 6);
                const float2 g = ((const float2*)srcRow)[lane];
                float* dst = outv + ((size_t)(unsigned)r << 6) + (lane << 1);
                unsafeAtomicAdd(dst,     v * g.x);        // global_atomic_add_f32, no return
                unsafeAtomicAdd(dst + 1, v * g.y);
            }
            buf ^= 1;
        }
    }

    // generic tail (unused when NNZ % TILE == 0, kept for correctness)
    const int tailStart = ntiles * TILE;
    if (tailStart < nnz) {
        const int gw = (int)blockIdx.x * (SPMM_BLOCK >> 5) + wid;
        const int nw = (int)gridDim.x * (SPMM_BLOCK >> 5);
        for (int e = tailStart + gw; e < nnz; e += nw) {
            const int r = __builtin_amdgcn_readfirstlane(rows[e]);
            const int c = __builtin_amdgcn_readfirstlane(cols[e]);
            const float v = __int_as_float(
                __builtin_amdgcn_readfirstlane(__float_as_int(vals[e])));
            const float* __restrict__ srcRow = ego + ((size_t)(unsigned)c << 6);
            const float2 g = ((const float2*)srcRow)[lane];
            float* dst = outv + ((size_t)(unsigned)r << 6) + (lane << 1);
            unsafeAtomicAdd(dst,     v * g.x);
            unsafeAtomicAdd(dst + 1, v * g.y);
        }
    }
}

// ---------------- elementwise helpers (float4 vectorized) ----------------
__global__ void lg_init(const float* __restrict__ user, const float* __restrict__ item,
                        float* __restrict__ ego, float* __restrict__ acc,
                        long long userElems, long long totalElems) {
    const long long n4 = totalElems >> 2;
    const long long u4 = userElems >> 2;
    const float4* uf = (const float4*)user;
    const float4* itf = (const float4*)item;
    float4* e4 = (float4*)ego;
    float4* a4 = (float4*)acc;
    for (long long i = (long long)blockIdx.x * blockDim.x + threadIdx.x;
         i < n4; i += (long long)gridDim.x * blockDim.x) {
        const float4 v = (i < u4) ? uf[i] : itf[i - u4];
        e4[i] = v;
        a4[i] = v;     // acc starts as layer-0 embedding
    }
}

__global__ void lg_zero(float* __restrict__ p, long long totalElems) {
    const long long n4 = totalElems >> 2;
    float4* p4 = (float4*)p;
    const float4 z = make_float4(0.f, 0.f, 0.f, 0.f);
    for (long long i = (long long)blockIdx.x * blockDim.x + threadIdx.x;
         i < n4; i += (long long)gridDim.x * blockDim.x)
        p4[i] = z;
}

__global__ void lg_accum_and_clear(float* __restrict__ acc, const float* __restrict__ layer,
                                   float* __restrict__ toClear, long long totalElems) {
    const long long n4 = totalElems >> 2;
    float4* a4 = (float4*)acc;
    const float4* l4 = (const float4*)layer;
    float4* c4 = (float4*)toClear;
    const float4 z = make_float4(0.f, 0.f, 0.f, 0.f);
    for (long long i = (long long)blockIdx.x * blockDim.x + threadIdx.x;
         i < n4; i += (long long)gridDim.x * blockDim.x) {
        float4 a = a4[i];
        const float4 l = l4[i];
        a.x += l.x; a.y += l.y; a.z += l.z; a.w += l.w;
        a4[i] = a;
        c4[i] = z;     // next layer's scatter target must start at zero
    }
}

__global__ void lg_scale(float* __restrict__ p, float s, long long totalElems) {
    const long long n4 = totalElems >> 2;
    float4* p4 = (float4*)p;
    for (long long i = (long long)blockIdx.x * blockDim.x + threadIdx.x;
         i < n4; i += (long long)gridDim.x * blockDim.x) {
        float4 v = p4[i];
        v.x *= s; v.y *= s; v.z *= s; v.w *= s;
        p4[i] = v;
    }
}

extern "C" void kernel_launch(void* const* d_in, const int* in_sizes, int n_in,
                              void* d_out, int out_size, void* d_ws, size_t ws_size,
                              hipStream_t stream) {
    const float* user = (const float*)d_in[0];
    const float* item = (const float*)d_in[1];
    const int*   rows = (const int*)d_in[2];
    const int*   cols = (const int*)d_in[3];
    const float* vals = (const float*)d_in[4];
    float* out = (float*)d_out;

    const long long userElems = in_sizes[0];
    const long long itemElems = in_sizes[1];
    const int nnz = in_sizes[2];
    const long long total = userElems + itemElems;   // N_NODES * 64

    float* A = (float*)d_ws;        // ego ping
    float* B = A + total;           // ego pong (2 x 76.8 MB in d_ws)

    // acc lives directly in d_out (output layout == [N_NODES, 64] flat)
    lg_init<<<EW_GRID, EW_BLOCK, 0, stream>>>(user, item, A, out, userElems, total);
    lg_zero<<<EW_GRID, EW_BLOCK, 0, stream>>>(B, total);

    for (int layer = 0; layer < 3; ++layer) {
        lg_spmm<<<SPMM_GRID, SPMM_BLOCK, 0, stream>>>(rows, cols, vals, A, B, nnz);
        lg_accum_and_clear<<<EW_GRID, EW_BLOCK, 0, stream>>>(out, B, A, total);
        float* tmp = A; A = B; B = tmp;   // B's result becomes next ego; cleared A is next target
    }

    lg_scale<<<EW_GRID, EW_BLOCK, 0, stream>>>(out, 0.25f, total);
}